// SharedGCNEncoder_69681549410755
// MI455X (gfx1250) — compile-verified
//
#include <hip/hip_runtime.h>

// ---------------------------------------------------------------------------
// 2-layer GCN for MI455X (gfx1250, wave32).
//   h  = relu( Dinv (A+I) Dinv (x W1) + b1 )
//   z  =       Dinv (A+I) Dinv (h W2) + b2
// GEMMs use V_WMMA_F32_16X16X4_F32 (full fp32 precision matrix pipe).
// Message passing is edge-parallel with global f32 atomics (L2-resident:
// the 25.6MB feature matrix fits in the 192MB L2).
// ---------------------------------------------------------------------------

typedef __attribute__((ext_vector_type(2))) float v2f;
typedef __attribute__((ext_vector_type(8))) float v8f;

#define N_NODES 50000
#define N_EDGES 800000
#define F_IN    128
#define F_HID   128
#define F_OUT   64

// ---------------- degree / normalization ----------------------------------

__global__ __launch_bounds__(256) void init_deg_kernel(float* __restrict__ deg) {
    int i = blockIdx.x * blockDim.x + threadIdx.x;
    if (i < N_NODES) deg[i] = 1.0f;            // self-loop contributes 1
}

__global__ __launch_bounds__(256) void accum_deg_kernel(const int* __restrict__ dst,
                                                        float* __restrict__ deg) {
    int e = blockIdx.x * blockDim.x + threadIdx.x;
    if (e < N_EDGES) atomicAdd(&deg[dst[e]], 1.0f);
}

__global__ __launch_bounds__(256) void dinv_kernel(float* __restrict__ deg_to_dinv) {
    int i = blockIdx.x * blockDim.x + threadIdx.x;
    if (i < N_NODES) deg_to_dinv[i] = rsqrtf(deg_to_dinv[i]);  // deg >= 1 always
}

// ---------------- WMMA fp32 GEMM: H[MxF] = X[MxK] @ W[KxF] ----------------
// One wave computes one 16x16 output tile, stepping K by 4 per WMMA.
// A (16x4 f32) lane layout: lanes 0-15 hold {K=k0,k0+1}, lanes 16-31 {k0+2,k0+3},
//   row M = lane&15.  B (4x16 f32): VGPR0 = rows k0 / k0+2, VGPR1 = k0+1 / k0+3,
//   col N = lane&15.  C/D: VGPR v -> row v (lanes 0-15) / v+8 (lanes 16-31).

__global__ __launch_bounds__(256) void gemm_wmma_f32(const float* __restrict__ X,
                                                     const float* __restrict__ W,
                                                     float* __restrict__ H,
                                                     int n_rows, int K, int F) {
    const int lane  = threadIdx.x & 31;
    const int wave  = blockIdx.x * (blockDim.x >> 5) + (threadIdx.x >> 5);
    const int tiles_x = F >> 4;
    const int row0 = (wave / tiles_x) << 4;
    const int col0 = (wave % tiles_x) << 4;
    if (row0 >= n_rows) return;

    const int half = lane >> 4;    // 0: lanes 0-15, 1: lanes 16-31
    const int l16  = lane & 15;

    const float* __restrict__ xrow = X + (size_t)(row0 + l16) * K;

    v8f c = {};
    for (int k0 = 0; k0 < K; k0 += 4) {
        const int ka = k0 + (half << 1);
        v2f a, b;
        a.x = xrow[ka + 0];
        a.y = xrow[ka + 1];
        b.x = W[(size_t)(ka + 0) * F + col0 + l16];
        b.y = W[(size_t)(ka + 1) * F + col0 + l16];
        c = __builtin_amdgcn_wmma_f32_16x16x4_f32(
                /*neg_a=*/false, a, /*neg_b=*/false, b,
                /*c_mod=*/(short)0, c, /*reuse_a=*/false, /*reuse_b=*/false);
    }

    #pragma unroll
    for (int v = 0; v < 8; ++v) {
        const int m = v + (half << 3);
        H[(size_t)(row0 + m) * F + col0 + l16] = c[v];
    }
}

// ---------------- zero a float buffer --------------------------------------

__global__ __launch_bounds__(256) void zero_kernel(float* __restrict__ p, size_t n) {
    size_t i = (size_t)blockIdx.x * blockDim.x + threadIdx.x;
    size_t stride = (size_t)gridDim.x * blockDim.x;
    for (; i < n; i += stride) p[i] = 0.0f;
}

// ---------------- edge-parallel normalized scatter-add ---------------------
// agg[dst[e]][f] += h[src[e]][f] * dinv[src]*dinv[dst]
// Thread i -> (edge e = i>>lgF, feature f = i & (F-1)): feature dim is the
// fast dim so 512B/256B atomic bursts per edge are fully coalesced.

__global__ __launch_bounds__(256) void scatter_kernel(const float* __restrict__ H,
                                                      const int* __restrict__ src,
                                                      const int* __restrict__ dst,
                                                      const float* __restrict__ dinv,
                                                      float* __restrict__ agg,
                                                      int lgF) {
    const int F = 1 << lgF;
    const size_t total  = (size_t)N_EDGES << lgF;
    size_t i = (size_t)blockIdx.x * blockDim.x + threadIdx.x;
    const size_t stride = (size_t)gridDim.x * blockDim.x;
    for (; i < total; i += stride) {
        const int e = (int)(i >> lgF);
        const int f = (int)(i & (F - 1));
        const int s = src[e];
        const int d = dst[e];
        const float nrm = dinv[s] * dinv[d];
        atomicAdd(&agg[((size_t)d << lgF) + f], H[((size_t)s << lgF) + f] * nrm);
    }
}

// ---------------- epilogue: self-loop term + bias (+ relu) -----------------
// out = [relu]( agg + dinv^2 * h + bias )

__global__ __launch_bounds__(256) void epilogue_kernel(const float* __restrict__ agg,
                                                       const float* __restrict__ H,
                                                       const float* __restrict__ dinv,
                                                       const float* __restrict__ bias,
                                                       float* __restrict__ out,
                                                       int lgF, int do_relu) {
    const int F = 1 << lgF;
    const size_t total = (size_t)N_NODES << lgF;
    size_t i = (size_t)blockIdx.x * blockDim.x + threadIdx.x;
    if (i >= total) return;
    const int node = (int)(i >> lgF);
    const int f    = (int)(i & (F - 1));
    const float di = dinv[node];
    float v = agg[i] + di * di * H[i] + bias[f];
    if (do_relu) v = fmaxf(v, 0.0f);
    out[i] = v;
}

// ---------------------------------------------------------------------------

extern "C" void kernel_launch(void* const* d_in, const int* in_sizes, int n_in,
                              void* d_out, int out_size, void* d_ws, size_t ws_size,
                              hipStream_t stream) {
    (void)in_sizes; (void)n_in; (void)out_size; (void)ws_size;

    const float* x   = (const float*)d_in[0];
    const int*   ei  = (const int*)d_in[1];    // [2, E]: row0 = src, row1 = dst
    const float* W1  = (const float*)d_in[2];
    const float* b1  = (const float*)d_in[3];
    const float* W2  = (const float*)d_in[4];
    const float* b2  = (const float*)d_in[5];
    float*       out = (float*)d_out;

    const int* src = ei;
    const int* dst = ei + N_EDGES;

    // Workspace layout (floats): dinv | h (N*128) | agg1 (N*128) | agg2 (N*64)
    float* ws   = (float*)d_ws;
    float* dinv = ws;
    float* h    = dinv + N_NODES;
    float* agg1 = h    + (size_t)N_NODES * F_HID;
    float* agg2 = agg1 + (size_t)N_NODES * F_HID;

    const int T = 256;

    // ---- normalization coefficients ----
    init_deg_kernel<<<(N_NODES + T - 1) / T, T, 0, stream>>>(dinv);
    accum_deg_kernel<<<(N_EDGES + T - 1) / T, T, 0, stream>>>(dst, dinv);
    dinv_kernel<<<(N_NODES + T - 1) / T, T, 0, stream>>>(dinv);

    // ---- layer 1: h = x @ W1 (WMMA), propagate, relu ----
    {
        const int tiles = (N_NODES / 16) * (F_HID / 16);     // 3125 * 8 = 25000 waves
        const int waves_per_block = T / 32;                  // 8
        gemm_wmma_f32<<<(tiles + waves_per_block - 1) / waves_per_block, T, 0, stream>>>(
            x, W1, h, N_NODES, F_IN, F_HID);

        zero_kernel<<<2048, T, 0, stream>>>(agg1, (size_t)N_NODES * F_HID);
        scatter_kernel<<<4096, T, 0, stream>>>(h, src, dst, dinv, agg1, /*lgF=*/7);

        const size_t total = (size_t)N_NODES * F_HID;
        epilogue_kernel<<<(unsigned)((total + T - 1) / T), T, 0, stream>>>(
            agg1, h, dinv, b1, /*out=*/agg1, /*lgF=*/7, /*relu=*/1);   // agg1 becomes h1
    }

    // ---- layer 2: h2 = h1 @ W2 (WMMA), propagate ----
    {
        const int tiles = (N_NODES / 16) * (F_OUT / 16);     // 3125 * 4 = 12500 waves
        const int waves_per_block = T / 32;
        gemm_wmma_f32<<<(tiles + waves_per_block - 1) / waves_per_block, T, 0, stream>>>(
            agg1, W2, /*H=*/h, N_NODES, F_HID, F_OUT);       // reuse h buffer as h2

        zero_kernel<<<2048, T, 0, stream>>>(agg2, (size_t)N_NODES * F_OUT);
        scatter_kernel<<<4096, T, 0, stream>>>(h, src, dst, dinv, agg2, /*lgF=*/6);

        const size_t total = (size_t)N_NODES * F_OUT;
        epilogue_kernel<<<(unsigned)((total + T - 1) / T), T, 0, stream>>>(
            agg2, h, dinv, b2, out, /*lgF=*/6, /*relu=*/0);
    }
}